// GCNBondLayer_31671088841192
// MI455X (gfx1250) — compile-verified
//
#include <hip/hip_runtime.h>
#include <hip/hip_bf16.h>

typedef __attribute__((ext_vector_type(2))) float v2f;
typedef __attribute__((ext_vector_type(8))) float v8f;

#define WAVES_PER_BLOCK 8
#define P1_BLOCKS 1024
#define BN_EPS 1e-5f

// D = A(16x4) x B(4x16) + C, fp32; 4 k-steps give K=16.
__device__ __forceinline__ v8f wmma4(v2f a, v2f b, v8f c) {
    return __builtin_amdgcn_wmma_f32_16x16x4_f32(false, a, false, b, (short)0, c,
                                                 false, false);
}

// B-fragment for weight matrix W (row-major [16][16], used as A @ W^T).
// b[j] for k-step kk = W[n][4*kk + 2*h + j]  -> aligned float2.
__device__ __forceinline__ void load_wfrag(const float* __restrict__ W, int n, int h,
                                           v2f bw[4]) {
#pragma unroll
    for (int kk = 0; kk < 4; ++kk) {
        const float2* p = reinterpret_cast<const float2*>(W + n * 16 + 4 * kk + 2 * h);
        float2 v = *p;
        bw[kk].x = v.x;
        bw[kk].y = v.y;
    }
}

// A-fragment from a row-major [*,16] matrix, row index pre-clamped by caller.
__device__ __forceinline__ void load_afrag(const float* __restrict__ M, int row, int h,
                                           v2f a[4]) {
    const float* rp = M + (size_t)row * 16;
#pragma unroll
    for (int kk = 0; kk < 4; ++kk) {
        const float2* p = reinterpret_cast<const float2*>(rp + 4 * kk + 2 * h);
        float2 v = *p;
        a[kk].x = v.x;
        a[kk].y = v.y;
    }
}

// ---------------- Kernel 1: per-node transform  y2 = x@w2^T + b2,  y3 = x@w3^T + b3
__global__ __launch_bounds__(256) void gcn_node_transform(
    const float* __restrict__ x, const float* __restrict__ w2,
    const float* __restrict__ b2, const float* __restrict__ w3,
    const float* __restrict__ b3, float* __restrict__ y2, float* __restrict__ y3,
    int N) {
    const int lane = threadIdx.x & 31;
    const int wave = threadIdx.x >> 5;
    const int h = lane >> 4;
    const int n = lane & 15;

    v2f bw2[4], bw3[4];
    load_wfrag(w2, n, h, bw2);
    load_wfrag(w3, n, h, bw3);
    const float bias2 = b2[n];
    const float bias3 = b3[n];

    const int ntiles = (N + 15) >> 4;
    const int gwave = blockIdx.x * WAVES_PER_BLOCK + wave;
    const int nwaves = gridDim.x * WAVES_PER_BLOCK;

    for (int t = gwave; t < ntiles; t += nwaves) {
        const int base = t << 4;
        int arow = base + n;
        if (arow >= N) arow = N - 1;  // clamp loads; EXEC stays all-ones for WMMA
        v2f a[4];
        load_afrag(x, arow, h, a);

        v8f acc2 = {}, acc3 = {};
#pragma unroll
        for (int kk = 0; kk < 4; ++kk) {
            acc2 = wmma4(a[kk], bw2[kk], acc2);
            acc3 = wmma4(a[kk], bw3[kk], acc3);
        }

        const int rowbase = base + 8 * h;
        float* p2 = y2 + (size_t)rowbase * 16 + n;
        float* p3 = y3 + (size_t)rowbase * 16 + n;
        if (base + 16 <= N) {  // full tile: branch-free, immediate-offset stores
#pragma unroll
            for (int r = 0; r < 8; ++r) {
                p2[r * 16] = acc2[r] + bias2;
                p3[r * 16] = acc3[r] + bias3;
            }
        } else {
#pragma unroll
            for (int r = 0; r < 8; ++r) {
                if (rowbase + r < N) {
                    p2[r * 16] = acc2[r] + bias2;
                    p3[r * 16] = acc3[r] + bias3;
                }
            }
        }
    }
}

// ---------------- Kernel 2: pass 1 over edges — compute e, accumulate per-feature
// sum / sumsq into deterministic per-block partials.
__global__ __launch_bounds__(256) void gcn_edge_pass1(
    const float* __restrict__ edge_attr, const int* __restrict__ edge_index,
    const float* __restrict__ w1, const float* __restrict__ b1,
    const float* __restrict__ y2, const float* __restrict__ y3,
    float* __restrict__ partials, int E) {
    __shared__ float sm[WAVES_PER_BLOCK * 16 * 2];

    const int lane = threadIdx.x & 31;
    const int wave = threadIdx.x >> 5;
    const int h = lane >> 4;
    const int n = lane & 15;

    v2f bw1[4];
    load_wfrag(w1, n, h, bw1);
    const float bias1 = b1[n];

    float s = 0.f, sq = 0.f;

    const int ntiles = (E + 15) >> 4;
    const int gwave = blockIdx.x * WAVES_PER_BLOCK + wave;
    const int nwaves = gridDim.x * WAVES_PER_BLOCK;

    for (int t = gwave; t < ntiles; t += nwaves) {
        const int base = t << 4;

        // prefetch next grid-stride tile of the streaming input
        const int nb = base + (nwaves << 4);
        if (nb + 15 < E) __builtin_prefetch(edge_attr + (size_t)(nb + n) * 16, 0, 0);

        int arow = base + n;
        if (arow >= E) arow = E - 1;
        v2f a[4];
        load_afrag(edge_attr, arow, h, a);

        v8f acc = {};
#pragma unroll
        for (int kk = 0; kk < 4; ++kk) acc = wmma4(a[kk], bw1[kk], acc);

        const int rowbase = base + 8 * h;
        const int* ip0 = edge_index + rowbase;
        const int* ip1 = edge_index + E + rowbase;
        if (base + 16 <= E) {  // full tile fast path
#pragma unroll
            for (int r = 0; r < 8; ++r) {
                const int sidx = ip0[r];
                const int didx = ip1[r];
                const float v = acc[r] + bias1 + y2[(size_t)sidx * 16 + n] +
                                y3[(size_t)didx * 16 + n];
                s += v;
                sq += v * v;
            }
        } else {
#pragma unroll
            for (int r = 0; r < 8; ++r) {
                int g = rowbase + r;
                const bool ok = (g < E);
                if (!ok) g = E - 1;
                const int sidx = edge_index[g];
                const int didx = edge_index[E + g];
                const float v = acc[r] + bias1 + y2[(size_t)sidx * 16 + n] +
                                y3[(size_t)didx * 16 + n];
                if (ok) {
                    s += v;
                    sq += v * v;
                }
            }
        }
    }

    // reduce lane halves (feature n duplicated between h=0 and h=1)
    s += __shfl_xor(s, 16, 32);
    sq += __shfl_xor(sq, 16, 32);
    if (h == 0) {
        sm[wave * 16 + n] = s;
        sm[WAVES_PER_BLOCK * 16 + wave * 16 + n] = sq;
    }
    __syncthreads();
    if (threadIdx.x < 16) {
        float ts = 0.f, tq = 0.f;
#pragma unroll
        for (int w = 0; w < WAVES_PER_BLOCK; ++w) {
            ts += sm[w * 16 + threadIdx.x];
            tq += sm[WAVES_PER_BLOCK * 16 + w * 16 + threadIdx.x];
        }
        partials[blockIdx.x * 32 + threadIdx.x] = ts;
        partials[blockIdx.x * 32 + 16 + threadIdx.x] = tq;
    }
}

// ---------------- Kernel 3: finalize BN stats -> scale/shift (32 floats)
__global__ void gcn_stats(const float* __restrict__ partials, int nblocks,
                          const float* __restrict__ gamma,
                          const float* __restrict__ beta, float* __restrict__ stats,
                          int E) {
    __shared__ float red[32];
    const int j = threadIdx.x;  // blockDim == 32
    float s = 0.f;
    for (int b = 0; b < nblocks; ++b) s += partials[b * 32 + j];
    red[j] = s;
    __syncthreads();
    if (j < 16) {
        const float invE = 1.0f / (float)E;
        const float mean = red[j] * invE;
        const float var = red[16 + j] * invE - mean * mean;
        const float scale = gamma[j] * rsqrtf(var + BN_EPS);
        stats[j] = scale;
        stats[16 + j] = beta[j] - mean * scale;
    }
}

// ---------------- Kernel 4: pass 2 — recompute e, normalize, ReLU, residual, store.
__global__ __launch_bounds__(256) void gcn_edge_pass2(
    const float* __restrict__ edge_attr, const int* __restrict__ edge_index,
    const float* __restrict__ w1, const float* __restrict__ b1,
    const float* __restrict__ y2, const float* __restrict__ y3,
    const float* __restrict__ stats, float* __restrict__ out, int E) {
    const int lane = threadIdx.x & 31;
    const int wave = threadIdx.x >> 5;
    const int h = lane >> 4;
    const int n = lane & 15;

    v2f bw1[4];
    load_wfrag(w1, n, h, bw1);
    const float bias1 = b1[n];
    const float scale = stats[n];
    const float shift = stats[16 + n];

    const int ntiles = (E + 15) >> 4;
    const int gwave = blockIdx.x * WAVES_PER_BLOCK + wave;
    const int nwaves = gridDim.x * WAVES_PER_BLOCK;

    for (int t = gwave; t < ntiles; t += nwaves) {
        const int base = t << 4;

        const int nb = base + (nwaves << 4);
        if (nb + 15 < E) __builtin_prefetch(edge_attr + (size_t)(nb + n) * 16, 0, 0);

        int arow = base + n;
        if (arow >= E) arow = E - 1;
        v2f a[4];
        load_afrag(edge_attr, arow, h, a);

        v8f acc = {};
#pragma unroll
        for (int kk = 0; kk < 4; ++kk) acc = wmma4(a[kk], bw1[kk], acc);

        const int rowbase = base + 8 * h;
        const int* ip0 = edge_index + rowbase;
        const int* ip1 = edge_index + E + rowbase;
        const float* ep = edge_attr + (size_t)rowbase * 16 + n;
        float* op = out + (size_t)rowbase * 16 + n;

        if (base + 16 <= E) {  // full tile fast path, immediate-offset ld/st
#pragma unroll
            for (int r = 0; r < 8; ++r) {
                const int sidx = ip0[r];
                const int didx = ip1[r];
                const float v = acc[r] + bias1 + y2[(size_t)sidx * 16 + n] +
                                y3[(size_t)didx * 16 + n];
                const float nrm = fmaxf(v * scale + shift, 0.0f);
                op[r * 16] = ep[r * 16] + nrm;
            }
        } else {
#pragma unroll
            for (int r = 0; r < 8; ++r) {
                int g = rowbase + r;
                const bool ok = (g < E);
                if (!ok) g = E - 1;
                const int sidx = edge_index[g];
                const int didx = edge_index[E + g];
                const float v = acc[r] + bias1 + y2[(size_t)sidx * 16 + n] +
                                y3[(size_t)didx * 16 + n];
                const float nrm = fmaxf(v * scale + shift, 0.0f);
                if (ok) op[r * 16] = ep[r * 16] + nrm;
            }
        }
    }
}

extern "C" void kernel_launch(void* const* d_in, const int* in_sizes, int n_in,
                              void* d_out, int out_size, void* d_ws, size_t ws_size,
                              hipStream_t stream) {
    const float* x = (const float*)d_in[0];
    const int* edge_index = (const int*)d_in[1];
    const float* edge_attr = (const float*)d_in[2];
    const float* w1 = (const float*)d_in[3];
    const float* b1 = (const float*)d_in[4];
    const float* w2 = (const float*)d_in[5];
    const float* b2 = (const float*)d_in[6];
    const float* w3 = (const float*)d_in[7];
    const float* b3 = (const float*)d_in[8];
    const float* gamma = (const float*)d_in[9];
    const float* beta = (const float*)d_in[10];
    float* out = (float*)d_out;

    const int N = in_sizes[0] / 16;
    const int E = in_sizes[1] / 2;

    // workspace layout
    float* y2 = (float*)d_ws;
    float* y3 = y2 + (size_t)N * 16;
    float* partials = y3 + (size_t)N * 16;
    float* stats = partials + (size_t)P1_BLOCKS * 32;

    const int node_tiles = (N + 15) / 16;
    int nb_node = (node_tiles + WAVES_PER_BLOCK - 1) / WAVES_PER_BLOCK;
    if (nb_node > 512) nb_node = 512;
    if (nb_node < 1) nb_node = 1;

    gcn_node_transform<<<nb_node, 256, 0, stream>>>(x, w2, b2, w3, b3, y2, y3, N);
    gcn_edge_pass1<<<P1_BLOCKS, 256, 0, stream>>>(edge_attr, edge_index, w1, b1, y2,
                                                  y3, partials, E);
    gcn_stats<<<1, 32, 0, stream>>>(partials, P1_BLOCKS, gamma, beta, stats, E);
    gcn_edge_pass2<<<P1_BLOCKS, 256, 0, stream>>>(edge_attr, edge_index, w1, b1, y2,
                                                  y3, stats, out, E);
}